// CheckpointFirstDivergenceLoss_48344151884226
// MI455X (gfx1250) — compile-verified
//
#include <hip/hip_runtime.h>
#include <hip/hip_bf16.h>

typedef __attribute__((ext_vector_type(2))) float v2f;
typedef __attribute__((ext_vector_type(8))) float v8f;

// Full-precision 32-lane sum using the fp32 WMMA pipe.
// A (16x4 f32) layout: lanes 0-15 -> {K0,K1}, lanes 16-31 -> {K2,K3} (ISA 7.12.2).
// We place the lane partial in the first A slot, zero in the second, B = ones.
// D[m][n] = sum_k A[m][k] = x[m] + x[m+16]; D vgpr r: lanes 0-15 hold S_r,
// lanes 16-31 hold S_{8+r}. Summing the 8 D vgprs in-lane gives the half-sums;
// one xor-16 shuffle completes the full wave sum (EXEC must be all ones here).
__device__ __forceinline__ float wave_sum32(float x) {
    v2f a; a[0] = x;    a[1] = 0.0f;
    v2f b; b[0] = 1.0f; b[1] = 1.0f;
    v8f c = {};
    v8f d = __builtin_amdgcn_wmma_f32_16x16x4_f32(
        /*neg_a=*/false, a, /*neg_b=*/false, b,
        /*c_mod=*/(short)0, c, /*reuse_a=*/false, /*reuse_b=*/false);
    float t = d[0] + d[1] + d[2] + d[3] + d[4] + d[5] + d[6] + d[7];
    t += __shfl_xor(t, 16, 32);
    return t;
}

__device__ __forceinline__ float bce_term(float s, float lab) {
    float lp  = fmaxf(logf(s),     -100.0f);
    float l1  = fmaxf(log1pf(-s),  -100.0f);
    return -(lab * lp + (1.0f - lab) * l1);
}

__global__ void cfd_init_ws(double* ws) {
    if (threadIdx.x < 2) ws[threadIdx.x] = 0.0;
}

__global__ __launch_bounds__(256) void cfd_loss_main(
        const float* __restrict__ scores,
        const float* __restrict__ labels,
        const int*   __restrict__ t_star,
        double*      __restrict__ ws,
        int N4, int P) {
    const int tid      = blockIdx.x * blockDim.x + threadIdx.x;
    const int nthreads = gridDim.x * blockDim.x;
    const float4* __restrict__ s4 = (const float4*)scores;
    const float4* __restrict__ l4 = (const float4*)labels;

    float bce  = 0.0f;
    float rank = 0.0f;

    // Streaming BCE pass: 128-bit loads, memory-bound.
    for (int i = tid; i < N4; i += nthreads) {
        __builtin_prefetch(&s4[i + nthreads], 0, 0);   // global_prefetch_b8
        __builtin_prefetch(&l4[i + nthreads], 0, 0);
        float4 s = s4[i];
        float4 l = l4[i];
        bce += bce_term(s.x, l.x);
        bce += bce_term(s.y, l.y);
        bce += bce_term(s.z, l.z);
        bce += bce_term(s.w, l.w);
    }

    // Ranking pass: scores just streamed -> 33.5 MB fits easily in 192 MB L2.
    for (int p = tid; p < P; p += nthreads) {
        int   base = p << 5;             // 32 elements per pair
        int   t    = t_star[base];       // replicated per pair; read first slot
        float ref  = scores[base + t];
        float dev  = scores[base + 16 + t];
        // -log_sigmoid(ref - dev) == softplus(dev - ref), computed stably
        float y = dev - ref;
        rank += fmaxf(y, 0.0f) + log1pf(expf(-fabsf(y)));
    }

    // Wave reduction on the matrix pipe (all 32 lanes active here).
    float wr = wave_sum32(rank);
    float wb = wave_sum32(bce);

    __shared__ float sR[8];
    __shared__ float sB[8];
    const int wave = threadIdx.x >> 5;
    const int lane = threadIdx.x & 31;
    if (lane == 0) { sR[wave] = wr; sB[wave] = wb; }
    __syncthreads();

    if (threadIdx.x == 0) {
        double r = 0.0, b = 0.0;
        #pragma unroll
        for (int w = 0; w < 8; ++w) { r += (double)sR[w]; b += (double)sB[w]; }
        atomicAdd(&ws[0], r);   // global_atomic_add_f64
        atomicAdd(&ws[1], b);
    }
}

__global__ void cfd_finalize(const double* __restrict__ ws,
                             float* __restrict__ out, int P, int N) {
    if (blockIdx.x == 0 && threadIdx.x == 0) {
        out[0] = (float)(ws[0] / (double)P);   // ranking_loss
        out[1] = (float)(ws[1] / (double)N);   // bce_loss
    }
}

extern "C" void kernel_launch(void* const* d_in, const int* in_sizes, int n_in,
                              void* d_out, int out_size, void* d_ws, size_t ws_size,
                              hipStream_t stream) {
    const float* scores = (const float*)d_in[0];
    const float* labels = (const float*)d_in[1];
    const int*   t_star = (const int*)d_in[5];

    const int N  = in_sizes[0];       // 8,388,608
    const int P  = N / 32;            // 262,144 pairs
    const int N4 = N / 4;

    double* ws  = (double*)d_ws;
    float*  out = (float*)d_out;

    cfd_init_ws<<<1, 32, 0, stream>>>(ws);

    const int blocks  = 2048;         // 8 waves/block; 4 float4 iters/thread
    const int threads = 256;
    cfd_loss_main<<<blocks, threads, 0, stream>>>(scores, labels, t_star, ws, N4, P);

    cfd_finalize<<<1, 32, 0, stream>>>(ws, out, P, N);
}